// VectorizedGAT_37606733644294
// MI455X (gfx1250) — compile-verified
//
#include <hip/hip_runtime.h>
#include <hip/hip_bf16.h>

// ---------------------------------------------------------------------------
// GATConv (concat heads) for MI455X / gfx1250.
//   N=50000 nodes, F_IN=128, HEADS=4, D=32 (HD=128), E=1.6M edges.
// Phase 1: h = x @ W via V_WMMA_F32_16X16X4_F32 (fp32-exact matrix pipe).
// Phase 2-5: edge softmax + scatter aggregation (HBM/atomic bound).
// ---------------------------------------------------------------------------

typedef __attribute__((ext_vector_type(2))) float v2f;
typedef __attribute__((ext_vector_type(8))) float v8f;

#define HD 128          // HEADS * OUT_PER_HEAD
#define NHEADS 4
#define DHEAD 32
#define NEG_SLOPE 0.2f

// ---------------- init: out = bias, mbits = 0 (== -inf key), denom = 0 -----
__global__ void __launch_bounds__(256)
gat_init(float* __restrict__ out, const float* __restrict__ bias,
         unsigned* __restrict__ mbits, float* __restrict__ denom, int N) {
  int gid = blockIdx.x * blockDim.x + threadIdx.x;
  if (gid < N * HD) out[gid] = bias[gid & (HD - 1)];
  if (gid < N * NHEADS) { mbits[gid] = 0u; denom[gid] = 0.0f; }
}

// ---------------- projection: h = x @ W via WMMA f32 16x16x4 ---------------
// One wave -> one 16x16 output tile. Block = 256 threads = 8 waves covering
// the full 128-wide output for 16 rows of x. K loop: 32 steps of K=4.
__global__ void __launch_bounds__(256)
gat_proj_wmma(const float* __restrict__ x, const float* __restrict__ W,
              float* __restrict__ h) {
  const int lane  = threadIdx.x & 31;
  const int wave  = threadIdx.x >> 5;      // 0..7 -> N tile
  const int half  = lane >> 4;             // 0/1 (K sub-group)
  const int l16   = lane & 15;
  const int mBase = blockIdx.x * 16;
  const int nBase = wave * 16;
  const int nc    = nBase + l16;           // B column this lane loads
  const float* xrow = x + (size_t)(mBase + l16) * HD;

  v8f c = {};
#pragma unroll
  for (int k0 = 0; k0 < 128; k0 += 4) {
    const int ka = k0 + 2 * half;
    // A fragment: A[m=l16][ka], A[m=l16][ka+1]  (contiguous -> b64 load)
    v2f a = *(const v2f*)(xrow + ka);
    // B fragment: B[ka][nc], B[ka+1][nc]
    v2f b;
    b.x = W[(size_t)ka * HD + nc];
    b.y = W[(size_t)(ka + 1) * HD + nc];
    // (neg_a, A, neg_b, B, c_mod, C, reuse_a, reuse_b)
    c = __builtin_amdgcn_wmma_f32_16x16x4_f32(false, a, false, b,
                                              (short)0, c, false, false);
  }
#pragma unroll
  for (int j = 0; j < 8; ++j) {
    const int row = mBase + j + 8 * half;  // D VGPR j: M=j (lanes 0-15), M=j+8
    h[(size_t)row * HD + nBase + l16] = c[j];
  }
}

// -------- scalar tail for M rows not covered by full 16-row tiles ----------
__global__ void gat_proj_tail(const float* __restrict__ x,
                              const float* __restrict__ W,
                              float* __restrict__ h, int startRow, int nRows) {
  int gid = blockIdx.x * blockDim.x + threadIdx.x;
  if (gid >= nRows * HD) return;
  int r = startRow + gid / HD;
  int cidx = gid & (HD - 1);
  float acc = 0.0f;
#pragma unroll 4
  for (int k = 0; k < 128; ++k) acc += x[(size_t)r * HD + k] * W[(size_t)k * HD + cidx];
  h[(size_t)r * HD + cidx] = acc;
}

// ---------------- per-node attention logits: es/ed[n,h] --------------------
// One wave per (node, head); lane = feature within head; wave reduce.
__global__ void __launch_bounds__(256)
gat_node_logits(const float* __restrict__ h, const float* __restrict__ a_src,
                const float* __restrict__ a_dst, float* __restrict__ es,
                float* __restrict__ ed, int N) {
  int gid  = blockIdx.x * blockDim.x + threadIdx.x;
  int pair = gid >> 5;                       // node*NHEADS + head (wave-uniform)
  if (pair >= N * NHEADS) return;
  int lane = gid & 31;
  int node = pair >> 2, head = pair & 3;
  float hv = h[(size_t)node * HD + head * DHEAD + lane];
  float s = hv * a_src[head * DHEAD + lane];
  float d = hv * a_dst[head * DHEAD + lane];
#pragma unroll
  for (int off = 16; off > 0; off >>= 1) {
    s += __shfl_down(s, off, 32);
    d += __shfl_down(d, off, 32);
  }
  if (lane == 0) { es[pair] = s; ed[pair] = d; }
}

// order-preserving monotone float->uint key (for atomic segment-max)
__device__ __forceinline__ unsigned f32_key(float v) {
  unsigned b = __float_as_uint(v);
  return (b & 0x80000000u) ? ~b : (b | 0x80000000u);
}
__device__ __forceinline__ float f32_unkey(unsigned u) {
  // u==0 => no edges (== -inf sentinel); reference clamps non-finite max to 0
  if (u == 0u) return 0.0f;
  return (u & 0x80000000u) ? __uint_as_float(u & 0x7fffffffu)
                           : __uint_as_float(~u);
}

// ---------------- per-edge logits + segment max over destinations ----------
__global__ void __launch_bounds__(256)
gat_edge_logits(const int* __restrict__ esrc, const int* __restrict__ edst,
                const float* __restrict__ es, const float* __restrict__ ed,
                float* __restrict__ ee, unsigned* __restrict__ mbits, int E) {
  int gid = blockIdx.x * blockDim.x + threadIdx.x;
  if (gid >= E * NHEADS) return;
  int e = gid >> 2, head = gid & 3;
  int s = esrc[e], d = edst[e];
  float v = es[s * NHEADS + head] + ed[d * NHEADS + head];
  v = (v > 0.0f) ? v : NEG_SLOPE * v;        // LeakyReLU
  ee[gid] = v;
  atomicMax(mbits + d * NHEADS + head, f32_key(v));
}

// ---------------- alpha = exp(e - m[dst]); denom = segment_sum -------------
__global__ void __launch_bounds__(256)
gat_alpha(const int* __restrict__ edst, float* __restrict__ ee,
          const unsigned* __restrict__ mbits, float* __restrict__ denom, int E) {
  int gid = blockIdx.x * blockDim.x + threadIdx.x;
  if (gid >= E * NHEADS) return;
  int e = gid >> 2, head = gid & 3;
  int d = edst[e];
  float mm = f32_unkey(mbits[d * NHEADS + head]);
  float a = expf(ee[gid] - mm);
  ee[gid] = a;
  atomicAdd(denom + d * NHEADS + head, a);
}

// ---------------- reciprocal of denominator (in place) ---------------------
__global__ void gat_rdenom(float* __restrict__ denom, int N) {
  int gid = blockIdx.x * blockDim.x + threadIdx.x;
  if (gid >= N * NHEADS) return;
  float v = denom[gid];
  denom[gid] = (v > 0.0f) ? 1.0f / v : 1.0f;
}

// ---------------- weighted aggregation (memory/atomic bound) ---------------
// Thread handles 4 consecutive features of one edge: float4 gather of h[src],
// 4x global_atomic_add_f32 onto out[dst].
__global__ void __launch_bounds__(256)
gat_aggregate(const int* __restrict__ esrc, const int* __restrict__ edst,
              const float* __restrict__ alpha, const float* __restrict__ rden,
              const float* __restrict__ h, float* __restrict__ out, int E) {
  int gid = blockIdx.x * blockDim.x + threadIdx.x;   // over E * 32
  if (gid >= E * 32) return;
  int e  = gid >> 5;
  int q  = gid & 31;
  int f0 = q << 2;                  // feature start 0..124
  int head = q >> 3;                // f0 / 32
  int s = esrc[e], d = edst[e];
  float w = alpha[(size_t)e * NHEADS + head] * rden[d * NHEADS + head];
  const float4 hv = *(const float4*)(h + (size_t)s * HD + f0);
  float* o = out + (size_t)d * HD + f0;
  atomicAdd(o + 0, w * hv.x);
  atomicAdd(o + 1, w * hv.y);
  atomicAdd(o + 2, w * hv.z);
  atomicAdd(o + 3, w * hv.w);
}

// ---------------------------------------------------------------------------
extern "C" void kernel_launch(void* const* d_in, const int* in_sizes, int n_in,
                              void* d_out, int out_size, void* d_ws, size_t ws_size,
                              hipStream_t stream) {
  const float* x     = (const float*)d_in[0];
  const float* W     = (const float*)d_in[1];
  const float* a_src = (const float*)d_in[2];
  const float* a_dst = (const float*)d_in[3];
  const float* bias  = (const float*)d_in[4];
  const int*   esrc  = (const int*)d_in[5];
  const int*   edst  = (const int*)d_in[6];
  float* out = (float*)d_out;

  const int N = in_sizes[0] / HD;     // 50000
  const int E = in_sizes[5];          // 1600000

  // workspace layout (all segments 16B-aligned)
  char* ws = (char*)d_ws;
  float*    h     = (float*)ws;    ws += (size_t)N * HD * sizeof(float);
  float*    es    = (float*)ws;    ws += (size_t)N * NHEADS * sizeof(float);
  float*    ed    = (float*)ws;    ws += (size_t)N * NHEADS * sizeof(float);
  unsigned* mbits = (unsigned*)ws; ws += (size_t)N * NHEADS * sizeof(unsigned);
  float*    denom = (float*)ws;    ws += (size_t)N * NHEADS * sizeof(float);
  float*    ee    = (float*)ws;    // E * NHEADS floats (logits -> alpha)

  const int B = 256;

  // 1) out = bias, reset softmax state
  gat_init<<<(N * HD + B - 1) / B, B, 0, stream>>>(out, bias, mbits, denom, N);

  // 2) h = x @ W  (WMMA f32 16x16x4; 8 waves/block cover the 128-wide output)
  const int mTiles = N / 16;
  if (mTiles > 0)
    gat_proj_wmma<<<mTiles, 256, 0, stream>>>(x, W, h);
  const int rem = N - mTiles * 16;
  if (rem > 0)
    gat_proj_tail<<<(rem * HD + B - 1) / B, B, 0, stream>>>(x, W, h, mTiles * 16, rem);

  // 3) per-node attention logits
  gat_node_logits<<<(N * HD + B - 1) / B, B, 0, stream>>>(h, a_src, a_dst, es, ed, N);

  // 4) per-edge LeakyReLU logits + segment max
  gat_edge_logits<<<(E * NHEADS + B - 1) / B, B, 0, stream>>>(esrc, edst, es, ed, ee, mbits, E);

  // 5) alpha = exp(e - m[dst]); segment sum of denominators
  gat_alpha<<<(E * NHEADS + B - 1) / B, B, 0, stream>>>(edst, ee, mbits, denom, E);

  // 6) denom -> reciprocal (with reference's denom>0 guard)
  gat_rdenom<<<(N * NHEADS + B - 1) / B, B, 0, stream>>>(denom, N);

  // 7) weighted scatter aggregation
  gat_aggregate<<<(E * 32 + B - 1) / B, B, 0, stream>>>(esrc, edst, ee, denom, h, out, E);
}